// LSTM_NATTEN_79104707658572
// MI455X (gfx1250) — compile-verified
//
#include <hip/hip_runtime.h>
#include <hip/hip_bf16.h>
#include <math.h>

typedef __attribute__((ext_vector_type(2))) float v2f;
typedef __attribute__((ext_vector_type(8))) float v8f;

// Problem constants (from reference)
constexpr int CB  = 2;     // batch
constexpr int CT  = 128;   // time
constexpr int CF  = 88;    // freq
constexpr int CH  = 64;    // hidden / n_unit
constexpr int CNH = 4;     // heads
constexpr int CW  = 7;     // window
constexpr int CS  = CB * CF;        // 176 sequences
constexpr int CM  = CB * CT * CF;   // 22528 grid rows

__device__ __forceinline__ float sigf(float x) {
  return 1.0f / (1.0f + __expf(-x));
}

// ---------------------------------------------------------------------------
// (B, T*F, H) -> (B*F, T, H) sequence layout
// ---------------------------------------------------------------------------
__global__ void reorder_kernel(const float* __restrict__ x, float* __restrict__ seq) {
  int gid = blockIdx.x * blockDim.x + threadIdx.x;
  if (gid >= CS * CT * CH) return;
  int c = gid & 63;
  int r = gid >> 6;
  int t = r & (CT - 1);
  int s = r >> 7;            // CT = 128
  int b = s / CF, f = s % CF;
  seq[gid] = x[((size_t)(b * (CT * CF) + t * CF + f) << 6) + c];
}

// ---------------------------------------------------------------------------
// One BiLSTM layer. grid = (11, 2): blockIdx.x = 16-seq tile, blockIdx.y = dir.
// 256 threads = 8 waves; wave w owns gate columns [16w, 16w+16).
// Per step: gates(16x128) = [X(t)|h](16x96) @ Wt(96x128) via 24 f32 WMMAs/wave.
// ---------------------------------------------------------------------------
__global__ __launch_bounds__(256)
void lstm_layer_kernel(const float* __restrict__ in_seq,   // (176,128,64)
                       float* __restrict__ out,            // seq or grid layout
                       const float* __restrict__ w_ih_all,  // (2,2,128,64)
                       const float* __restrict__ w_hh_all,  // (2,2,128,32)
                       const float* __restrict__ b_ih_all,  // (2,2,128)
                       const float* __restrict__ b_hh_all,  // (2,2,128)
                       int layer, int out_grid) {
  const int dir  = blockIdx.y;
  const int seq_base = blockIdx.x * 16;
  const int tid  = threadIdx.x;
  const int lane = tid & 31;
  const int wave = tid >> 5;
  const int n0   = wave * 16;

  const float* wih = w_ih_all + (size_t)(layer * 2 + dir) * 128 * 64;
  const float* whh = w_hh_all + (size_t)(layer * 2 + dir) * 128 * 32;
  const float* bih = b_ih_all + (size_t)(layer * 2 + dir) * 128;
  const float* bhh = b_hh_all + (size_t)(layer * 2 + dir) * 128;

  __shared__ float xbuf[16][64];   // X(t) tile
  __shared__ float hbuf[16][32];   // current hidden state
  __shared__ float gbuf[16][128];  // gate pre-activations

  // --- preload B fragments (weights, K=96 over 24 steps of 4) into registers
  const int ncol  = n0 + (lane & 15);
  const int khalf = (lane < 16) ? 0 : 2;
  float wb0[24], wb1[24];
#pragma unroll
  for (int kk = 0; kk < 24; ++kk) {
    int k0 = kk * 4 + khalf;
    int k1 = k0 + 1;
    wb0[kk] = (k0 < 64) ? wih[ncol * 64 + k0] : whh[ncol * 32 + (k0 - 64)];
    wb1[kk] = (k1 < 64) ? wih[ncol * 64 + k1] : whh[ncol * 32 + (k1 - 64)];
  }

  // --- elementwise thread mapping: element e = (s,u); this thread owns
  //     (s0,u) and (s0+8,u)
  const int u  = tid & 31;
  const int s0 = tid >> 5;   // 0..7
  const int s1 = s0 + 8;
  const float bi = bih[u]      + bhh[u];
  const float bf = bih[32 + u] + bhh[32 + u];
  const float bg = bih[64 + u] + bhh[64 + u];
  const float bo = bih[96 + u] + bhh[96 + u];

  // zero state
  ((float*)hbuf)[tid]       = 0.0f;
  ((float*)hbuf)[tid + 256] = 0.0f;
  float c0 = 0.0f, c1 = 0.0f;
  __syncthreads();

  const int arow = lane & 15;

  for (int step = 0; step < CT; ++step) {
    const int t = dir ? (CT - 1 - step) : step;

    // stage X(t): 16x64 floats, 4 per thread
#pragma unroll
    for (int ii = 0; ii < 4; ++ii) {
      int e = tid + ii * 256;
      int s = e >> 6, k = e & 63;
      xbuf[s][k] = in_seq[(size_t)(seq_base + s) * CT * 64 + (size_t)t * 64 + k];
    }
    __syncthreads();

    // fused gate GEMM: acc(16x16) over K=96
    v8f acc = {};
#pragma unroll
    for (int kk = 0; kk < 24; ++kk) {
      int k0 = kk * 4 + khalf;
      v2f a, b;
      a.x = (k0     < 64) ? xbuf[arow][k0]     : hbuf[arow][k0 - 64];
      a.y = (k0 + 1 < 64) ? xbuf[arow][k0 + 1] : hbuf[arow][k0 + 1 - 64];
      b.x = wb0[kk];
      b.y = wb1[kk];
      acc = __builtin_amdgcn_wmma_f32_16x16x4_f32(false, a, false, b,
                                                  (short)0, acc, false, false);
    }

    // spill C fragment: VGPR r -> row r (lanes<16) / r+8 (lanes>=16)
    {
      const int grow = (lane < 16) ? 0 : 8;
      const int gcol = n0 + (lane & 15);
#pragma unroll
      for (int r = 0; r < 8; ++r) gbuf[grow + r][gcol] = acc[r];
    }
    __syncthreads();

    // elementwise LSTM cell update for (s0,u) and (s1,u)
    {
      float gi = gbuf[s0][u] + bi;
      float gf = gbuf[s0][32 + u] + bf;
      float gg = gbuf[s0][64 + u] + bg;
      float go = gbuf[s0][96 + u] + bo;
      c0 = sigf(gf) * c0 + sigf(gi) * tanhf(gg);
      float h = sigf(go) * tanhf(c0);
      hbuf[s0][u] = h;
      int sg = seq_base + s0;
      size_t idx;
      if (out_grid) {
        int bb = sg / CF, ff = sg % CF;
        idx = (((size_t)bb * CT + t) * CF + ff) * 64 + dir * 32 + u;
      } else {
        idx = ((size_t)sg * CT + t) * 64 + dir * 32 + u;
      }
      out[idx] = h;
    }
    {
      float gi = gbuf[s1][u] + bi;
      float gf = gbuf[s1][32 + u] + bf;
      float gg = gbuf[s1][64 + u] + bg;
      float go = gbuf[s1][96 + u] + bo;
      c1 = sigf(gf) * c1 + sigf(gi) * tanhf(gg);
      float h = sigf(go) * tanhf(c1);
      hbuf[s1][u] = h;
      int sg = seq_base + s1;
      size_t idx;
      if (out_grid) {
        int bb = sg / CF, ff = sg % CF;
        idx = (((size_t)bb * CT + t) * CF + ff) * 64 + dir * 32 + u;
      } else {
        idx = ((size_t)sg * CT + t) * 64 + dir * 32 + u;
      }
      out[idx] = h;
    }
    __syncthreads();
  }
}

// ---------------------------------------------------------------------------
// h_e = h + t_emb[t[b]]  (broadcast over (i,j))
// ---------------------------------------------------------------------------
__global__ void add_emb_kernel(const float* __restrict__ h,
                               const float* __restrict__ emb,  // (100,64) for this layer
                               const int* __restrict__ tq,
                               float* __restrict__ he) {
  int gid = blockIdx.x * blockDim.x + threadIdx.x;
  if (gid >= CM * CH) return;
  int c = gid & 63;
  int row = gid >> 6;
  int b = row / (CT * CF);
  he[gid] = h[gid] + emb[tq[b] * 64 + c];
}

// ---------------------------------------------------------------------------
// Generic f32 WMMA GEMM: C = A(MxK) @ B(KxN) [+ bias] [+ residual]
// One wave per 16x16 output tile; 8 waves per block; K compile-time, fully
// unrolled (16 straight-line WMMAs for K=64). A fragment loaded as b64.
// ---------------------------------------------------------------------------
template <int K>
__global__ __launch_bounds__(256)
void wmma_gemm_kernel(const float* __restrict__ A, const float* __restrict__ Bm,
                      const float* __restrict__ bias, const float* __restrict__ R,
                      float* __restrict__ C, int M, int N) {
  const int lane = threadIdx.x & 31;
  const int wave = threadIdx.x >> 5;
  const int mtiles = M >> 4, ntiles = N >> 4;
  const int tile = blockIdx.x * 8 + wave;
  if (tile >= mtiles * ntiles) return;   // wave-uniform
  const int mt = tile / ntiles, nt = tile % ntiles;

  const int arow  = mt * 16 + (lane & 15);
  const int bcol  = nt * 16 + (lane & 15);
  const int khalf = (lane < 16) ? 0 : 2;

  const float* Arow = A + (size_t)arow * K + khalf;   // 8B aligned (khalf even)

  v8f acc = {};
#pragma unroll
  for (int k = 0; k < K; k += 4) {
    v2f a = *reinterpret_cast<const v2f*>(Arow + k);  // one global_load_b64
    v2f b;
    b.x = Bm[(size_t)(k + khalf) * N + bcol];
    b.y = Bm[(size_t)(k + khalf + 1) * N + bcol];
    acc = __builtin_amdgcn_wmma_f32_16x16x4_f32(false, a, false, b,
                                                (short)0, acc, false, false);
  }

  const int crow0 = mt * 16 + ((lane < 16) ? 0 : 8);
  const int ccol  = nt * 16 + (lane & 15);
  const float bv  = bias ? bias[ccol] : 0.0f;
#pragma unroll
  for (int r = 0; r < 8; ++r) {
    size_t idx = (size_t)(crow0 + r) * N + ccol;
    float v = acc[r] + bv;
    if (R) v += R[idx];
    C[idx] = v;
  }
}

// ---------------------------------------------------------------------------
// Neighborhood attention 7x7: one thread per (b, head, i, j).
// qkv row layout: [3][4 heads][16]; all q/k/v slices 16B aligned -> float4.
// ---------------------------------------------------------------------------
__global__ __launch_bounds__(256)
void natten_kernel(const float* __restrict__ qkv,
                   const float* __restrict__ rpb,   // (4,13,13) for this layer
                   float* __restrict__ out) {
  int gid = blockIdx.x * blockDim.x + threadIdx.x;
  if (gid >= CB * CNH * CT * CF) return;
  int j = gid % CF;  int tmp = gid / CF;
  int i = tmp % CT;  tmp /= CT;
  int hd = tmp % CNH;
  int b  = tmp / CNH;

  const size_t row = (size_t)(b * CT + i) * CF + j;
  const float4* qp = reinterpret_cast<const float4*>(qkv + row * 192 + hd * 16);
  float4 q[4];
#pragma unroll
  for (int d = 0; d < 4; ++d) {
    float4 v = qp[d];
    q[d] = make_float4(v.x * 0.25f, v.y * 0.25f, v.z * 0.25f, v.w * 0.25f);
  }

  int si = i - 3; if (si < 0) si = 0; if (si > CT - CW) si = CT - CW;
  int sj = j - 3; if (sj < 0) sj = 0; if (sj > CF - CW) sj = CF - CW;

  float logits[49];
  float mx = -1e30f;
  for (int ki = 0; ki < CW; ++ki) {
    for (int kj = 0; kj < CW; ++kj) {
      int ii = si + ki, jj = sj + kj;
      const float4* kp = reinterpret_cast<const float4*>(
          qkv + ((size_t)(b * CT + ii) * CF + jj) * 192 + 64 + hd * 16);
      float acc = 0.0f;
#pragma unroll
      for (int d = 0; d < 4; ++d) {
        float4 kv = kp[d];
        acc += q[d].x * kv.x + q[d].y * kv.y + q[d].z * kv.z + q[d].w * kv.w;
      }
      int ri = ii - i + 6, rj = jj - j + 6;
      acc += rpb[(hd * 13 + ri) * 13 + rj];
      logits[ki * CW + kj] = acc;
      mx = fmaxf(mx, acc);
    }
  }
  float ssum = 0.0f;
#pragma unroll
  for (int e = 0; e < 49; ++e) { float w = __expf(logits[e] - mx); logits[e] = w; ssum += w; }
  const float inv = 1.0f / ssum;

  float4 o[4];
#pragma unroll
  for (int d = 0; d < 4; ++d) o[d] = make_float4(0.f, 0.f, 0.f, 0.f);
  for (int ki = 0; ki < CW; ++ki) {
    for (int kj = 0; kj < CW; ++kj) {
      int ii = si + ki, jj = sj + kj;
      const float4* vp = reinterpret_cast<const float4*>(
          qkv + ((size_t)(b * CT + ii) * CF + jj) * 192 + 128 + hd * 16);
      float w = logits[ki * CW + kj] * inv;
#pragma unroll
      for (int d = 0; d < 4; ++d) {
        float4 vv = vp[d];
        o[d].x += w * vv.x; o[d].y += w * vv.y;
        o[d].z += w * vv.z; o[d].w += w * vv.w;
      }
    }
  }
  float4* op = reinterpret_cast<float4*>(out + row * 64 + hd * 16);
#pragma unroll
  for (int d = 0; d < 4; ++d) op[d] = o[d];
}

// ---------------------------------------------------------------------------
extern "C" void kernel_launch(void* const* d_in, const int* in_sizes, int n_in,
                              void* d_out, int out_size, void* d_ws, size_t ws_size,
                              hipStream_t stream) {
  const float* x     = (const float*)d_in[0];
  const int*   tq    = (const int*)  d_in[1];
  const float* w_ih  = (const float*)d_in[2];
  const float* w_hh  = (const float*)d_in[3];
  const float* b_ih  = (const float*)d_in[4];
  const float* b_hh  = (const float*)d_in[5];
  const float* t_emb = (const float*)d_in[6];
  const float* w_qkv = (const float*)d_in[7];
  const float* b_qkv = (const float*)d_in[8];
  const float* rpb   = (const float*)d_in[9];
  const float* w_prj = (const float*)d_in[10];
  const float* b_prj = (const float*)d_in[11];
  float* outp = (float*)d_out;

  const size_t NGRID = (size_t)CM * CH;      // 1,441,792
  float* ws   = (float*)d_ws;
  float* buf0 = ws;                 // seq_x, later attention ping buffer
  float* buf1 = buf0 + NGRID;       // lstm layer0 out (seq), later h_e
  float* buf2 = buf1 + NGRID;       // grid h after BiLSTM
  float* qkvb = buf2 + NGRID;       // 22528 * 192
  float* attb = qkvb + (size_t)CM * 192;

  const int ELT_BLK = 256;
  const int n_elt = (int)NGRID;
  const int elt_grid = (n_elt + ELT_BLK - 1) / ELT_BLK;   // 5632

  // 1) reorder input to sequences
  reorder_kernel<<<elt_grid, ELT_BLK, 0, stream>>>(x, buf0);

  // 2) BiLSTM: layer 0 (seq layout out), layer 1 (grid layout out)
  lstm_layer_kernel<<<dim3(CS / 16, 2), 256, 0, stream>>>(
      buf0, buf1, w_ih, w_hh, b_ih, b_hh, 0, 0);
  lstm_layer_kernel<<<dim3(CS / 16, 2), 256, 0, stream>>>(
      buf1, buf2, w_ih, w_hh, b_ih, b_hh, 1, 1);

  // 3) two NATTEN layers with residual
  const int qkv_tiles  = (CM / 16) * (192 / 16);   // 16896
  const int proj_tiles = (CM / 16) * (64 / 16);    // 5632
  const int qkv_blocks  = (qkv_tiles  + 7) / 8;    // 2112
  const int proj_blocks = (proj_tiles + 7) / 8;    // 704
  const int nat_blocks  = (CB * CNH * CT * CF + 255) / 256;  // 352

  // layer 0: h_in = buf2, h_out = buf0
  add_emb_kernel<<<elt_grid, ELT_BLK, 0, stream>>>(buf2, t_emb + 0 * 100 * 64, tq, buf1);
  wmma_gemm_kernel<64><<<qkv_blocks, 256, 0, stream>>>(
      buf1, w_qkv + 0 * 64 * 192, b_qkv + 0 * 192, nullptr, qkvb, CM, 192);
  natten_kernel<<<nat_blocks, 256, 0, stream>>>(qkvb, rpb + 0 * CNH * 13 * 13, attb);
  wmma_gemm_kernel<64><<<proj_blocks, 256, 0, stream>>>(
      attb, w_prj + 0 * 64 * 64, b_prj + 0 * 64, buf2, buf0, CM, 64);

  // layer 1: h_in = buf0, h_out = d_out
  add_emb_kernel<<<elt_grid, ELT_BLK, 0, stream>>>(buf0, t_emb + 1 * 100 * 64, tq, buf1);
  wmma_gemm_kernel<64><<<qkv_blocks, 256, 0, stream>>>(
      buf1, w_qkv + 1 * 64 * 192, b_qkv + 1 * 192, nullptr, qkvb, CM, 192);
  natten_kernel<<<nat_blocks, 256, 0, stream>>>(qkvb, rpb + 1 * CNH * 13 * 13, attb);
  wmma_gemm_kernel<64><<<proj_blocks, 256, 0, stream>>>(
      attb, w_prj + 1 * 64 * 64, b_prj + 1 * 64, buf0, outp, CM, 64);
}